// HyboNet_17119739642318
// MI455X (gfx1250) — compile-verified
//
#include <hip/hip_runtime.h>
#include <hip/hip_bf16.h>
#include <math.h>

typedef __attribute__((ext_vector_type(2))) float v2f;
typedef __attribute__((ext_vector_type(8))) float v8f;
typedef __attribute__((ext_vector_type(4))) int   v4i;

#define EPSF 1e-8f

// ---- CDNA5 async global->LDS path (guarded so the file always compiles) ----
// Probe-revealed prototype: (v4i addrspace(1)*, v4i addrspace(3)*, imm int, imm int)
typedef __attribute__((address_space(1))) v4i* gv4i_t;
typedef __attribute__((address_space(3))) v4i* sv4i_t;

#if defined(__has_builtin)
#  if __has_builtin(__builtin_amdgcn_global_load_async_to_lds_b128)
#    define HAVE_ASYNC_LDS 1
#  else
#    define HAVE_ASYNC_LDS 0
#  endif
#  if __has_builtin(__builtin_amdgcn_s_wait_asynccnt)
#    define ASYNC_WAIT() __builtin_amdgcn_s_wait_asynccnt(0)
#  else
#    define ASYNC_WAIT() asm volatile("s_wait_asynccnt 0" ::: "memory")
#  endif
#else
#  define HAVE_ASYNC_LDS 0
#  define ASYNC_WAIT()
#endif

__device__ __forceinline__ float wave_reduce_sum(float v) {
  // wave32 butterfly reduction
  v += __shfl_xor(v, 16, 32);
  v += __shfl_xor(v, 8, 32);
  v += __shfl_xor(v, 4, 32);
  v += __shfl_xor(v, 2, 32);
  v += __shfl_xor(v, 1, 32);
  return v;
}

// ---------------------------------------------------------------------------
// 1) expmap0: (N,127) -> (N,128).  One wave per node.
// ---------------------------------------------------------------------------
__global__ void __launch_bounds__(256)
expmap0_kernel(const float* __restrict__ u, float* __restrict__ x, int Nn) {
  int lane = threadIdx.x & 31;
  int node = (int)((blockIdx.x * blockDim.x + threadIdx.x) >> 5);
  if (node >= Nn) return;
  const float* up = u + (size_t)node * 127;
  float vals[4];
  int cnt = 0;
  float ss = 0.f;
  for (int j = lane; j < 127; j += 32) { float t = up[j]; vals[cnt++] = t; ss += t * t; }
  ss = wave_reduce_sum(ss);
  float nrm = fmaxf(sqrtf(ss), EPSF);
  float sh = sinhf(nrm) / nrm;             // sqrt(C)=1
  float* xp = x + (size_t)node * 128;
  cnt = 0;
  for (int j = lane; j < 127; j += 32) xp[1 + j] = sh * vals[cnt++];
  if (lane == 0) xp[0] = coshf(nrm);
}

// ---------------------------------------------------------------------------
// 2) Lorentz linear: Y = lorentz_fix( (relu?)X @ W^T + b ).
//    Block = 256 threads = 8 waves.  The shared 16x128 X tile (contiguous
//    8KB) is staged once into LDS via async global->LDS copies (ASYNCcnt
//    path); each wave then computes a 16x16 tile with V_WMMA_F32_16X16X4_F32
//    (A frags from LDS, B frags from W in global — each W element is read
//    once per block).  Result staged to LDS for the fused per-row postprocess.
// ---------------------------------------------------------------------------
__global__ void __launch_bounds__(256)
lorentz_linear_kernel(const float* __restrict__ X, const float* __restrict__ W,
                      const float* __restrict__ bias, const float* __restrict__ log_s,
                      float* __restrict__ Y, int relu_in, int Nn) {
  __shared__ float tA[16 * 128];   // staged input tile
  __shared__ float tY[16 * 128];   // GEMM result staging for postprocess
  int tid  = threadIdx.x;
  int lane = tid & 31;
  int wave = tid >> 5;
  int m0 = blockIdx.x * 16;        // Nn % 16 == 0 for this workload

  // ---- stage A tile (8KB contiguous) into LDS: 256 lanes x 16B x 2 -------
  {
    const float* src = X + (size_t)m0 * 128;
#if HAVE_ASYNC_LDS
    __builtin_amdgcn_global_load_async_to_lds_b128(
        (gv4i_t)(src + tid * 4), (sv4i_t)(&tA[tid * 4]), 0, 0);
    __builtin_amdgcn_global_load_async_to_lds_b128(
        (gv4i_t)(src + 1024 + tid * 4), (sv4i_t)(&tA[1024 + tid * 4]), 0, 0);
    ASYNC_WAIT();
#else
    ((float4*)tA)[tid]       = ((const float4*)src)[tid];
    ((float4*)tA)[tid + 256] = ((const float4*)src)[tid + 256];
#endif
  }
  __syncthreads();

  int n0 = wave * 16;
  int hl = lane & 15;
  int kb = (lane >> 4) << 1;                 // 0 (lanes 0-15) or 2 (lanes 16-31)
  const float* arow = &tA[hl * 128 + kb];                   // A frags from LDS
  const float* brow = W + (size_t)(n0 + hl) * 128 + kb;     // B frags from global

  v8f acc = {};
#pragma unroll 8
  for (int k0 = 0; k0 < 128; k0 += 4) {
    float a0 = arow[k0], a1 = arow[k0 + 1];
    if (relu_in) { a0 = fmaxf(a0, 0.f); a1 = fmaxf(a1, 0.f); }
    v2f a = {a0, a1};
    v2f b = {brow[k0], brow[k0 + 1]};
    acc = __builtin_amdgcn_wmma_f32_16x16x4_f32(false, a, false, b, (short)0, acc,
                                                false, false);
  }

  float bn = bias[n0 + hl];
#pragma unroll
  for (int v = 0; v < 8; ++v) {
    int mr = (lane < 16) ? v : v + 8;        // C/D layout: VGPR v -> rows v / v+8
    tY[mr * 128 + n0 + hl] = acc[v] + bn;
  }
  __syncthreads();

  // Postprocess: wave handles rows 2w and 2w+1; lane covers dims 4*lane..4*lane+3
  float smax = fminf(expf(log_s[0]), 10.0f);
  for (int r = wave * 2; r < wave * 2 + 2; ++r) {
    if (m0 + r >= Nn) break;
    const float* tr = &tY[r * 128];
    float e0 = tr[lane * 4 + 0];
    float e1 = tr[lane * 4 + 1];
    float e2 = tr[lane * 4 + 2];
    float e3 = tr[lane * 4 + 3];
    float ss = e1 * e1 + e2 * e2 + e3 * e3 + ((lane == 0) ? 0.f : e0 * e0); // skip col 0
    ss = wave_reduce_sum(ss);
    float y0 = tr[0];
    float time = smax / (1.f + expf(-y0)) + 1.5f;    // sqrt(C)+0.5 = 1.5
    float sq = fmaxf(ss, EPSF);
    float scale = (time * time - 1.0f) / sq;
    float fac = sqrtf(fmaxf(scale, EPSF));
    float* yp = Y + (size_t)(m0 + r) * 128;
    yp[lane * 4 + 0] = (lane == 0) ? time : e0 * fac;
    yp[lane * 4 + 1] = e1 * fac;
    yp[lane * 4 + 2] = e2 * fac;
    yp[lane * 4 + 3] = e3 * fac;
  }
}

// ---------------------------------------------------------------------------
// 3a) zero workspace (graph-capture-safe; no host memset needed)
// ---------------------------------------------------------------------------
__global__ void __launch_bounds__(256)
zero_kernel(float4* __restrict__ p, int n4) {
  int i = blockIdx.x * blockDim.x + threadIdx.x;
  if (i < n4) p[i] = make_float4(0.f, 0.f, 0.f, 0.f);
}

// ---------------------------------------------------------------------------
// 3b) edge scatter: S[row] += w * X[col].  One wave per edge; lane covers a
//     float4.  Prefetch the next edge's source row (wave-uniform, so no
//     divergence) to hide gather latency behind the L2 atomic stream.
// ---------------------------------------------------------------------------
__global__ void __launch_bounds__(256)
agg_scatter_kernel(const float* __restrict__ X, const float* __restrict__ ew,
                   const int* __restrict__ er, const int* __restrict__ ec,
                   float* __restrict__ S, int Ne) {
  int lane = threadIdx.x & 31;
  int gw = (int)((blockIdx.x * blockDim.x + threadIdx.x) >> 5);
  int nw = (int)((gridDim.x * blockDim.x) >> 5);
  for (int e = gw; e < Ne; e += nw) {
    int en = e + nw;
    if (en < Ne) __builtin_prefetch(X + (size_t)ec[en] * 128, 0, 0);
    int col = ec[e];
    int row = er[e];
    float w = ew[e];
    const float4* src = (const float4*)(X + (size_t)col * 128);
    float4 v = src[lane];
    float* dst = S + (size_t)row * 128 + lane * 4;
    atomicAdd(dst + 0, v.x * w);
    atomicAdd(dst + 1, v.y * w);
    atomicAdd(dst + 2, v.z * w);
    atomicAdd(dst + 3, v.w * w);
  }
}

// ---------------------------------------------------------------------------
// 3c) Lorentz renorm of aggregated sums (in place).  One wave per node.
// ---------------------------------------------------------------------------
__global__ void __launch_bounds__(256)
agg_norm_kernel(float* __restrict__ S, int Nn) {
  int lane = threadIdx.x & 31;
  int node = (int)((blockIdx.x * blockDim.x + threadIdx.x) >> 5);
  if (node >= Nn) return;
  float4* sp = (float4*)(S + (size_t)node * 128);
  float4 v = sp[lane];
  float ss = v.x * v.x + v.y * v.y + v.z * v.z + v.w * v.w;
  float tot = wave_reduce_sum(ss);
  float s0 = __shfl(v.x, 0, 32);
  float neg_inner = 2.f * s0 * s0 - tot;     // s0^2 - sum(space^2)
  float inv = 1.0f / sqrtf(fmaxf(fabsf(neg_inner), EPSF));   // *sqrt(C)=1
  v.x *= inv; v.y *= inv; v.z *= inv; v.w *= inv;
  sp[lane] = v;
}

// ---------------------------------------------------------------------------
// 4) logits = 2C + 2*(x_signed @ cls^T) + cls_bias.  WMMA; 40 cols -> 3 tiles.
//    Sign flip of column k=0 and column masking folded branchlessly into B
//    (keeps EXEC all-1s around the WMMAs per ISA §7.12).
// ---------------------------------------------------------------------------
__global__ void __launch_bounds__(96)
logits_kernel(const float* __restrict__ X, const float* __restrict__ cls,
              const float* __restrict__ cls_bias, float* __restrict__ out, int Nn) {
  int lane = threadIdx.x & 31;
  int wave = threadIdx.x >> 5;               // 0..2
  int m0 = blockIdx.x * 16;
  int n0 = wave * 16;
  int hl = lane & 15;
  int kb = (lane >> 4) << 1;
  int ncol = n0 + hl;
  float bmask = (ncol < 40) ? 1.f : 0.f;
  int ccol = (ncol < 40) ? ncol : 39;        // clamped address, masked value
  int arowi = m0 + hl; if (arowi >= Nn) arowi = Nn - 1;
  const float* arow = X + (size_t)arowi * 128 + kb;
  const float* brow = cls + (size_t)ccol * 128 + kb;

  v8f acc = {};
#pragma unroll 8
  for (int k0 = 0; k0 < 128; k0 += 4) {
    v2f a = {arow[k0], arow[k0 + 1]};
    float sgn = ((k0 + kb) == 0) ? -1.f : 1.f;   // x_signed: negate time column
    float b0 = brow[k0] * bmask * sgn;
    float b1 = brow[k0 + 1] * bmask;             // k0+kb+1 is never 0
    v2f b = {b0, b1};
    acc = __builtin_amdgcn_wmma_f32_16x16x4_f32(false, a, false, b, (short)0, acc,
                                                false, false);
  }

  if (ncol < 40) {
    float cb = cls_bias[ncol];
#pragma unroll
    for (int v = 0; v < 8; ++v) {
      int mr = (lane < 16) ? v : v + 8;
      if (m0 + mr < Nn)
        out[(size_t)(m0 + mr) * 40 + ncol] = 2.0f + 2.0f * acc[v] + cb;
    }
  }
}

// ---------------------------------------------------------------------------
// Host launcher
// ---------------------------------------------------------------------------
extern "C" void kernel_launch(void* const* d_in, const int* in_sizes, int n_in,
                              void* d_out, int out_size, void* d_ws, size_t ws_size,
                              hipStream_t stream) {
  const float* node_feat = (const float*)d_in[0];
  const float* W1        = (const float*)d_in[1];
  const float* b1        = (const float*)d_in[2];
  const float* s1        = (const float*)d_in[3];
  const float* W2        = (const float*)d_in[4];
  const float* b2        = (const float*)d_in[5];
  const float* s2        = (const float*)d_in[6];
  const float* cls       = (const float*)d_in[7];
  const float* cls_bias  = (const float*)d_in[8];
  const float* edge_w    = (const float*)d_in[9];
  const int*   edge_row  = (const int*)d_in[10];
  const int*   edge_col  = (const int*)d_in[11];
  float* out = (float*)d_out;

  const int Nn = in_sizes[0] / 127;          // 50000
  const int Ne = in_sizes[9];                // 800000

  float* bufA = (float*)d_ws;                          // 25.6 MB
  float* bufB = bufA + (size_t)Nn * 128;               // 25.6 MB

  const int nodeBlocks = (Nn + 7) / 8;                 // 8 waves/block, 1 wave/node
  const int mTiles     = (Nn + 15) / 16;
  const int zBlocks    = (Nn * 32 + 255) / 256;        // Nn*128/4 float4 elems

  // x = expmap0(node_feat)                    -> bufA
  expmap0_kernel<<<nodeBlocks, 256, 0, stream>>>(node_feat, bufA, Nn);
  // x = lorentz_linear(x, W1, b1, s1)         -> bufB
  lorentz_linear_kernel<<<mTiles, 256, 0, stream>>>(bufA, W1, b1, s1, bufB, 0, Nn);
  // x = lorentz_agg(x)                        -> bufA
  zero_kernel<<<zBlocks, 256, 0, stream>>>((float4*)bufA, Nn * 32);
  agg_scatter_kernel<<<2048, 256, 0, stream>>>(bufB, edge_w, edge_row, edge_col, bufA, Ne);
  agg_norm_kernel<<<nodeBlocks, 256, 0, stream>>>(bufA, Nn);
  // x = lorentz_linear(relu(x), W2, b2, s2)   -> bufB
  lorentz_linear_kernel<<<mTiles, 256, 0, stream>>>(bufA, W2, b2, s2, bufB, 1, Nn);
  // x = lorentz_agg(x)                        -> bufA
  zero_kernel<<<zBlocks, 256, 0, stream>>>((float4*)bufA, Nn * 32);
  agg_scatter_kernel<<<2048, 256, 0, stream>>>(bufB, edge_w, edge_row, edge_col, bufA, Ne);
  agg_norm_kernel<<<nodeBlocks, 256, 0, stream>>>(bufA, Nn);
  // logits                                    -> d_out
  logits_kernel<<<mTiles, 96, 0, stream>>>(bufA, cls, cls_bias, out, Nn);
}